// MPNN_87514253623371
// MI455X (gfx1250) — compile-verified
//
#include <hip/hip_runtime.h>

#define N_NODES  100000
#define N_EDGES  640000
#define NODE_DIM 128
#define HIDDEN   128

// padded LDS row stride: 132 floats = 528 B (16B aligned for float4 staging,
// 4-bank rotation per row -> conflict-free 16-lane row-parallel reads)
#define LDS_STRIDE 132

typedef __attribute__((ext_vector_type(2))) float v2f;
typedef __attribute__((ext_vector_type(8))) float v8f;

__device__ __forceinline__ void wait_async_zero() {
#if __has_builtin(__builtin_amdgcn_s_wait_asynccnt)
    __builtin_amdgcn_s_wait_asynccnt(0);
#else
    asm volatile("s_wait_asynccnt 0x0" ::: "memory");
#endif
}

// ---------------------------------------------------------------------------
// Kernel 1: He = H @ W_node^T + b_node   (WMMA f32 16x16x4, K=128 -> 32 steps)
// One block = 16 rows of H; 8 waves, each owns one 16-wide column tile.
// A-tile staged Global->LDS with the CDNA5 async copy path (ASYNCcnt),
// bypassing VGPRs entirely.
// ---------------------------------------------------------------------------
__global__ __launch_bounds__(256) void mpnn_encode_kernel(
    const float* __restrict__ H, const float* __restrict__ Wn,
    const float* __restrict__ bn, float* __restrict__ He)
{
    __shared__ float Atile[16 * LDS_STRIDE];

    const int block_row = blockIdx.x * 16;
    const int tid = threadIdx.x;

    // Stage 16x128 f32 A-tile into LDS: async global->LDS b128 copies
    // (one 16B chunk per lane per iteration; 512 chunks total).
    {
        const float4* src = (const float4*)(H + (size_t)block_row * NODE_DIM);
        #pragma unroll
        for (int i = 0; i < 2; ++i) {
            int c    = tid + i * 256;     // 0..511 float4 chunks
            int row  = c >> 5;            // /32  (32 float4 per row)
            int col4 = c & 31;
            const float4* gptr = src + row * 32 + col4;
            // low 32 bits of the generic LDS pointer == LDS byte offset
            unsigned lds_addr = (unsigned)(size_t)&Atile[row * LDS_STRIDE + col4 * 4];
            asm volatile("global_load_async_to_lds_b128 %0, %1, off"
                         :: "v"(lds_addr), "v"(gptr) : "memory");
        }
        wait_async_zero();
    }
    __syncthreads();

    const int wave  = tid >> 5;                 // 0..7 -> output column tile
    const int lane  = tid & 31;
    const int m     = lane & 15;                // A row within tile
    const int khalf = (lane >> 4) << 1;         // 0 (lanes 0-15) or 2 (16-31)
    const int ncol  = wave * 16 + (lane & 15);  // output column 0..127

    v8f acc = {};
    #pragma unroll
    for (int k = 0; k < NODE_DIM / 4; ++k) {
        int kb = k * 4 + khalf;
        // A 16x4: lane holds A[m][kb], A[m][kb+1]
        v2f a = *(const v2f*)&Atile[m * LDS_STRIDE + kb];
        // B 4x16: B[k][n] = W_node[n][k] (He = H * W^T). Mirror of A layout.
        v2f b = *(const v2f*)&Wn[(size_t)ncol * NODE_DIM + kb];
        acc = __builtin_amdgcn_wmma_f32_16x16x4_f32(
            /*neg_a=*/false, a, /*neg_b=*/false, b,
            /*c_mod=*/(short)0, acc, /*reuse_a=*/false, /*reuse_b=*/false);
    }

    // C/D layout: VGPR r -> M = r + (lane>=16 ? 8 : 0), N = lane&15
    const float bias  = bn[ncol];
    const int   rbase = (lane >> 4) << 3;   // 0 or 8
    #pragma unroll
    for (int r = 0; r < 8; ++r) {
        int row = block_row + rbase + r;
        He[(size_t)row * HIDDEN + ncol] = acc[r] + bias;
    }
}

// ---------------------------------------------------------------------------
// Kernel 2: zero the aggregation buffer (workspace is poisoned; must re-zero
// every call for determinism).
// ---------------------------------------------------------------------------
__global__ __launch_bounds__(256) void mpnn_zero_kernel(float* __restrict__ agg)
{
    size_t i = (size_t)blockIdx.x * blockDim.x + threadIdx.x;
    const size_t n4 = (size_t)N_NODES * (HIDDEN / 4);
    if (i < n4) {
        float4 z = {0.f, 0.f, 0.f, 0.f};
        ((float4*)agg)[i] = z;
    }
}

// ---------------------------------------------------------------------------
// Kernel 3: edge scatter.  One wave per edge: lanes read one float4 of the
// 128-wide feature row of He[u] (coalesced 512B), scale by norm, and
// scatter-add into agg[v] with no-return f32 atomics (L2-resident).
// ---------------------------------------------------------------------------
__global__ __launch_bounds__(256) void mpnn_scatter_kernel(
    const long long* __restrict__ eidx,    // [2, E] int64, row-major
    const float* __restrict__ nfac,        // [E]
    const float* __restrict__ He,          // [N, 128]
    float* __restrict__ agg)               // [N, 128]
{
    const int e    = blockIdx.x * 8 + (threadIdx.x >> 5);   // wave id = edge id
    const int lane = threadIdx.x & 31;
    if (e >= N_EDGES) return;

    const int   u  = (int)eidx[e];
    const int   v  = (int)eidx[(size_t)N_EDGES + e];
    const float nf = nfac[e];

    float4 w = ((const float4*)(He + (size_t)u * HIDDEN))[lane];
    float* dst = agg + (size_t)v * HIDDEN + lane * 4;
    unsafeAtomicAdd(dst + 0, nf * w.x);
    unsafeAtomicAdd(dst + 1, nf * w.y);
    unsafeAtomicAdd(dst + 2, nf * w.z);
    unsafeAtomicAdd(dst + 3, nf * w.w);
}

// ---------------------------------------------------------------------------
// Kernel 4: out = relu((agg + He) @ W_upd^T + b_upd)   (WMMA f32, fused)
// Staging computes He+agg on the fly, so it keeps the VGPR load path.
// ---------------------------------------------------------------------------
__global__ __launch_bounds__(256) void mpnn_update_kernel(
    const float* __restrict__ He, const float* __restrict__ agg,
    const float* __restrict__ Wu, const float* __restrict__ bu,
    float* __restrict__ out)
{
    __shared__ float Atile[16 * LDS_STRIDE];

    const int block_row = blockIdx.x * 16;
    const int tid = threadIdx.x;

    // Stage (He + agg) 16x128 tile into LDS.
    {
        const float4* s0 = (const float4*)(He  + (size_t)block_row * HIDDEN);
        const float4* s1 = (const float4*)(agg + (size_t)block_row * HIDDEN);
        #pragma unroll
        for (int i = 0; i < 2; ++i) {
            int c    = tid + i * 256;
            int row  = c >> 5;
            int col4 = c & 31;
            float4 a = s0[row * 32 + col4];
            float4 b = s1[row * 32 + col4];
            float4 v = {a.x + b.x, a.y + b.y, a.z + b.z, a.w + b.w};
            *(float4*)&Atile[row * LDS_STRIDE + col4 * 4] = v;
        }
    }
    __syncthreads();

    const int wave  = tid >> 5;
    const int lane  = tid & 31;
    const int m     = lane & 15;
    const int khalf = (lane >> 4) << 1;
    const int ncol  = wave * 16 + (lane & 15);

    v8f acc = {};
    #pragma unroll
    for (int k = 0; k < HIDDEN / 4; ++k) {
        int kb = k * 4 + khalf;
        v2f a = *(const v2f*)&Atile[m * LDS_STRIDE + kb];
        v2f b = *(const v2f*)&Wu[(size_t)ncol * HIDDEN + kb];
        acc = __builtin_amdgcn_wmma_f32_16x16x4_f32(
            false, a, false, b, (short)0, acc, false, false);
    }

    const float bias  = bu[ncol];
    const int   rbase = (lane >> 4) << 3;
    #pragma unroll
    for (int r = 0; r < 8; ++r) {
        int   row = block_row + rbase + r;
        float x   = acc[r] + bias;
        out[(size_t)row * HIDDEN + ncol] = x > 0.f ? x : 0.f;
    }
}

// ---------------------------------------------------------------------------
extern "C" void kernel_launch(void* const* d_in, const int* in_sizes, int n_in,
                              void* d_out, int out_size, void* d_ws, size_t ws_size,
                              hipStream_t stream)
{
    const float*     H    = (const float*)d_in[0];
    const long long* eidx = (const long long*)d_in[1];   // int64 edge_index [2,E]
    const float*     nfac = (const float*)d_in[2];
    const float*     Wn   = (const float*)d_in[3];
    const float*     bn   = (const float*)d_in[4];
    const float*     Wu   = (const float*)d_in[5];
    const float*     bu   = (const float*)d_in[6];
    float*           out  = (float*)d_out;

    float* He  = (float*)d_ws;                           // 51.2 MB
    float* agg = He + (size_t)N_NODES * HIDDEN;          // 51.2 MB

    // 1) node encoder GEMM (WMMA f32, async Global->LDS staging)
    mpnn_encode_kernel<<<N_NODES / 16, 256, 0, stream>>>(H, Wn, bn, He);

    // 2) zero aggregation buffer
    mpnn_zero_kernel<<<(N_NODES * (HIDDEN / 4) + 255) / 256, 256, 0, stream>>>(agg);

    // 3) edge gather + scale + scatter-add (one wave per edge)
    mpnn_scatter_kernel<<<N_EDGES / 8, 256, 0, stream>>>(eidx, nfac, He, agg);

    // 4) residual + update GEMM + bias + relu (WMMA f32, fused)
    mpnn_update_kernel<<<N_NODES / 16, 256, 0, stream>>>(He, agg, Wu, bu, out);
}